// MyLoss_59871844106287
// MI455X (gfx1250) — compile-verified
//
#include <hip/hip_runtime.h>
#include <hip/hip_bf16.h>

// ---------------------------------------------------------------------------
// YOLO-style loss, fp32, B*S = 4,194,304 cells x 8 channels (pred + target).
// Memory-bound streaming reduction: 256 MB single pass -> scalar.
//   Stage 1: grid-stride over cells, 5 partial sums per block -> workspace.
//   Stage 2: one block folds block partials and composes the final scalar.
// Cross-lane reduction uses V_WMMA_F32_16X16X4_F32 (A=ones) to fold the two
// wave32 half-waves in one matrix op at full f32 precision.
// ---------------------------------------------------------------------------

typedef __attribute__((ext_vector_type(4))) float f4;
typedef __attribute__((ext_vector_type(2))) float v2f;
typedef __attribute__((ext_vector_type(8))) float v8f;

#define NBLK 2048
#define NTHR 256
#define EPSF 1e-6f
#define THRELD 0.2f

// Sum v over all 32 lanes of a wave32. First WMMA_F32_16X16X4_F32 with A=ones,
// B = {v, 0}: D[m][n] = sum_k B[k][n] = v[n] + v[n+16] (n = lane & 15), since
// the zeroed B VGPR removes the other K contributions for any K<->VGPR map.
// Then 4 xor-shuffle adds fold the remaining 16 lanes.
__device__ __forceinline__ float wave_reduce_sum(float v) {
    v2f a; a[0] = 1.0f; a[1] = 1.0f;        // A = ones(16x4)
    v2f b; b[0] = v;    b[1] = 0.0f;        // B: lane value in one VGPR, other zero
    v8f c = {};
    v8f d = __builtin_amdgcn_wmma_f32_16x16x4_f32(
        /*neg_a=*/false, a, /*neg_b=*/false, b,
        /*c_mod=*/(short)0, c, /*reuse_a=*/false, /*reuse_b=*/false);
    float w = d[0];                          // w = v[lane&15] + v[(lane&15)+16]
    w += __shfl_xor(w, 8, 32);
    w += __shfl_xor(w, 4, 32);
    w += __shfl_xor(w, 2, 32);
    w += __shfl_xor(w, 1, 32);
    return w;
}

// Block-level reduce of 5 accumulators -> out5 (valid in thread 0 only).
__device__ __forceinline__ void block_reduce5(float* acc, float* out5) {
    __shared__ float sred[NTHR / 32][5];
    const int wave = threadIdx.x >> 5;
    const int lane = threadIdx.x & 31;
    float r[5];
#pragma unroll
    for (int j = 0; j < 5; ++j) r[j] = wave_reduce_sum(acc[j]);
    if (lane == 0) {
#pragma unroll
        for (int j = 0; j < 5; ++j) sred[wave][j] = r[j];
    }
    __syncthreads();
    if (threadIdx.x == 0) {
#pragma unroll
        for (int j = 0; j < 5; ++j) {
            float s = 0.0f;
            for (int w = 0; w < NTHR / 32; ++w) s += sred[w][j];
            out5[j] = s;
        }
    }
}

// acc indices: 0=nooobj, 1=class, 2=contain, 3=loc, 4=not_contain
__global__ void __launch_bounds__(NTHR) loss_stage1(
        const float* __restrict__ pred, const float* __restrict__ tgt,
        float* __restrict__ partials, long long ncells) {
    float acc[5] = {0.0f, 0.0f, 0.0f, 0.0f, 0.0f};

    const long long stride = (long long)gridDim.x * NTHR;
    for (long long c = (long long)blockIdx.x * NTHR + threadIdx.x;
         c < ncells; c += stride) {
        const f4* p4 = (const f4*)(pred + c * 8);
        const f4* t4 = (const f4*)(tgt + c * 8);
        // Non-temporal: 256 MB single-use stream, larger than the 192 MB L2.
        f4 pa = __builtin_nontemporal_load(p4);
        f4 pb = __builtin_nontemporal_load(p4 + 1);
        f4 ta = __builtin_nontemporal_load(t4);
        f4 tb = __builtin_nontemporal_load(t4 + 1);
        // channels: [0]=c1 [1]=w1 [2]=conf1 [3]=c2 | [4]=w2 [5]=conf2 [6..7]=cls

        const bool coo = ta[2] > THRELD;

        // no-object confidence loss (channels 2 and 5)
        const float d2 = pa[2] - ta[2];
        const float d5 = pb[1] - tb[1];
        acc[0] += coo ? 0.0f : (d2 * d2 + d5 * d5);

        // class loss (channels 6..7)
        const float d6 = pb[2] - tb[2];
        const float d7 = pb[3] - tb[3];
        acc[1] += coo ? (d6 * d6 + d7 * d7) : 0.0f;

        // 1-D IoU of both predicted boxes vs first target box
        const float tlo  = ta[0] - 0.5f * ta[1];
        const float thi  = ta[0] + 0.5f * ta[1];
        const float plo0 = pa[0] - 0.5f * pa[1];
        const float phi0 = pa[0] + 0.5f * pa[1];
        const float plo1 = pa[3] - 0.5f * pb[0];
        const float phi1 = pa[3] + 0.5f * pb[0];
        const float a2   = thi - tlo;
        float in0 = fminf(phi0, thi) - fmaxf(plo0, tlo);
        float in1 = fminf(phi1, thi) - fmaxf(plo1, tlo);
        in0 = fmaxf(in0, 0.0f);
        in1 = fmaxf(in1, 0.0f);
        const float iou0 = in0 / ((phi0 - plo0) + a2 - in0);
        const float iou1 = in1 / ((phi1 - plo1) + a2 - in1);
        const bool r1 = iou1 > iou0;   // argmax: index 1 only if strictly greater

        // responsible pred box: r1 ? (pa3,pb0,pb1) : (pa0,pa1,pa2)
        // responsible tgt  box: r1 ? (ta3,tb0,tb1) : (ta0,ta1,ta2)
        const float pr_c    = r1 ? pa[3] : pa[0];
        const float pr_conf = r1 ? pb[1] : pa[2];
        const float tr_c    = r1 ? ta[3] : ta[0];
        const float tr_conf = r1 ? tb[1] : ta[2];
        const float pn_conf = r1 ? pa[2] : pb[1];  // not-responsible pred conf

        const float dc = pr_conf - tr_conf;
        const float dl = pr_c - tr_c;
        acc[2] += coo ? dc * dc : 0.0f;
        acc[3] += coo ? dl * dl : 0.0f;
        acc[4] += coo ? pn_conf * pn_conf : 0.0f;
    }

    float out5[5];
    block_reduce5(acc, out5);
    if (threadIdx.x == 0) {
#pragma unroll
        for (int j = 0; j < 5; ++j)
            partials[(long long)blockIdx.x * 5 + j] = out5[j];
    }
}

__global__ void __launch_bounds__(NTHR) loss_stage2(
        const float* __restrict__ partials, float* __restrict__ out, int nblk) {
    float acc[5] = {0.0f, 0.0f, 0.0f, 0.0f, 0.0f};
    for (int b = threadIdx.x; b < nblk; b += NTHR) {
#pragma unroll
        for (int j = 0; j < 5; ++j) acc[j] += partials[b * 5 + j];
    }
    float s[5];
    block_reduce5(acc, s);
    if (threadIdx.x == 0) {
        const float nooobj  = s[0];
        const float cls     = s[1];
        const float contain = s[2];
        const float loc     = s[3];
        const float ncf     = s[4] + EPSF;   // not_contain_loss includes +EPS
        out[0] = loc + cls * contain + ncf + nooobj + cls + EPSF;
    }
}

extern "C" void kernel_launch(void* const* d_in, const int* in_sizes, int n_in,
                              void* d_out, int out_size, void* d_ws, size_t ws_size,
                              hipStream_t stream) {
    const float* pred = (const float*)d_in[0];
    const float* tgt  = (const float*)d_in[1];
    float* out        = (float*)d_out;
    float* partials   = (float*)d_ws;          // NBLK*5 floats = 40 KB

    const long long ncells = (long long)in_sizes[0] / 8;

    loss_stage1<<<NBLK, NTHR, 0, stream>>>(pred, tgt, partials, ncells);
    loss_stage2<<<1, NTHR, 0, stream>>>(partials, out, NBLK);
}